// GAT_65326452572361
// MI455X (gfx1250) — compile-verified
//
#include <hip/hip_runtime.h>
#include <hip/hip_bf16.h>

// ---------------- problem constants (match reference) ----------------
#define NNODES 50000
#define NEDGES 800000
#define FIN    256
#define NHEAD  4
#define HDIM   64
#define FOUT   256          // NHEAD*HDIM
#define NCLUS  16
#define NEGSLOPE 0.2f
#define WLDSTRIDE 260       // 256 + 4 pad floats: kills lane-half bank conflict

typedef __attribute__((ext_vector_type(2))) float v2f;
typedef __attribute__((ext_vector_type(8))) float v8f;

// flip(-inf): -inf = 0xFF800000, sign bit set -> ~u = 0x007FFFFF (host-usable constant)
static constexpr unsigned FLIP_NEG_INF = 0x007FFFFFu;

// ---------------- monotonic float<->u32 flip for atomic max ----------------
__device__ __forceinline__ unsigned flipf(float f) {
    unsigned u = __float_as_uint(f);
    return (u >> 31) ? ~u : (u | 0x80000000u);
}
__device__ __forceinline__ float unflipf(unsigned x) {
    return (x >> 31) ? __uint_as_float(x & 0x7FFFFFFFu) : __uint_as_float(~x);
}

// ---------------- fill kernels ----------------
__global__ void fill_f32(float* __restrict__ p, float v, int n) {
    int i = blockIdx.x * blockDim.x + threadIdx.x;
    if (i < n) p[i] = v;
}
__global__ void fill_u32(unsigned* __restrict__ p, unsigned v, int n) {
    int i = blockIdx.x * blockDim.x + threadIdx.x;
    if (i < n) p[i] = v;
}

// ---------------- fp32 WMMA GEMM: Out[nrows,256] = A[nrows,256] @ W[256,256] ----------------
// W staged fully in LDS (260KB padded). 8 waves/block; wave w owns cols [32w, 32w+32):
// 2 N-tiles -> 8 accumulators; A fragments software-pipelined one k-step ahead.
__global__ void __launch_bounds__(256) gemm_f32_wmma(const float* __restrict__ A,
                                                     const float* __restrict__ W,
                                                     float* __restrict__ Out,
                                                     int nrows, int mtiles) {
    extern __shared__ float ldsW[];   // [FIN][WLDSTRIDE]
    const int tid = threadIdx.x;

    // cooperative stage of W into LDS (float4 per thread, padded rows)
    for (int t = tid; t < (FIN * FOUT) / 4; t += 256) {
        const int flat = t * 4;
        const int k = flat >> 8;          // row (K)
        const int n = flat & 255;         // col (N), multiple of 4
        const float4 w4 = *(const float4*)(W + flat);
        *(float4*)(&ldsW[k * WLDSTRIDE + n]) = w4;
    }
    __syncthreads();

    const int wave = tid >> 5;
    const int lane = tid & 31;
    const int half = lane >> 4;           // 0: K=k0..k0+1, 1: K=k0+2..k0+3
    const int l15  = lane & 15;
    const int n0a  = wave * 32;
    const int n0b  = n0a + 16;
    const int koff = 2 * half;

    for (int mtile = blockIdx.x; mtile < mtiles; mtile += gridDim.x) {
        const int m0 = mtile * 64;
        int r0 = m0 +  0 + l15; if (r0 >= nrows) r0 = nrows - 1;
        int r1 = m0 + 16 + l15; if (r1 >= nrows) r1 = nrows - 1;
        int r2 = m0 + 32 + l15; if (r2 >= nrows) r2 = nrows - 1;
        int r3 = m0 + 48 + l15; if (r3 >= nrows) r3 = nrows - 1;
        const float* a0p = A + (size_t)r0 * FIN;
        const float* a1p = A + (size_t)r1 * FIN;
        const float* a2p = A + (size_t)r2 * FIN;
        const float* a3p = A + (size_t)r3 * FIN;

        v8f acc0a = {}, acc1a = {}, acc2a = {}, acc3a = {};
        v8f acc0b = {}, acc1b = {}, acc2b = {}, acc3b = {};

        // prologue: load k-step 0 fragments
        v2f a0c = *(const v2f*)(a0p + koff);
        v2f a1c = *(const v2f*)(a1p + koff);
        v2f a2c = *(const v2f*)(a2p + koff);
        v2f a3c = *(const v2f*)(a3p + koff);

        #pragma unroll 4
        for (int k0 = 0; k0 < FIN; k0 += 4) {
            const int kk = k0 + koff;
            // prefetch next k-step's A fragments (clamped: last iter reloads current)
            const int kp = (k0 + 4 < FIN) ? (kk + 4) : kk;
            v2f a0n = *(const v2f*)(a0p + kp);
            v2f a1n = *(const v2f*)(a1p + kp);
            v2f a2n = *(const v2f*)(a2p + kp);
            v2f a3n = *(const v2f*)(a3p + kp);

            v2f ba, bb;
            ba.x = ldsW[kk * WLDSTRIDE + n0a + l15];
            ba.y = ldsW[(kk + 1) * WLDSTRIDE + n0a + l15];
            bb.x = ldsW[kk * WLDSTRIDE + n0b + l15];
            bb.y = ldsW[(kk + 1) * WLDSTRIDE + n0b + l15];

            acc0a = __builtin_amdgcn_wmma_f32_16x16x4_f32(false, a0c, false, ba, (short)0, acc0a, false, false);
            acc0b = __builtin_amdgcn_wmma_f32_16x16x4_f32(false, a0c, false, bb, (short)0, acc0b, false, false);
            acc1a = __builtin_amdgcn_wmma_f32_16x16x4_f32(false, a1c, false, ba, (short)0, acc1a, false, false);
            acc1b = __builtin_amdgcn_wmma_f32_16x16x4_f32(false, a1c, false, bb, (short)0, acc1b, false, false);
            acc2a = __builtin_amdgcn_wmma_f32_16x16x4_f32(false, a2c, false, ba, (short)0, acc2a, false, false);
            acc2b = __builtin_amdgcn_wmma_f32_16x16x4_f32(false, a2c, false, bb, (short)0, acc2b, false, false);
            acc3a = __builtin_amdgcn_wmma_f32_16x16x4_f32(false, a3c, false, ba, (short)0, acc3a, false, false);
            acc3b = __builtin_amdgcn_wmma_f32_16x16x4_f32(false, a3c, false, bb, (short)0, acc3b, false, false);

            a0c = a0n; a1c = a1n; a2c = a2n; a3c = a3n;
        }

        // D layout: VGPR v, lanes 0-15 -> M=v, lanes 16-31 -> M=v+8; N = lane&15
        const int cola = n0a + l15;
        const int colb = n0b + l15;
        #pragma unroll
        for (int v = 0; v < 8; ++v) {
            const int mr = v + half * 8;
            int r;
            r = m0 +  0 + mr; if (r < nrows) { Out[(size_t)r * FOUT + cola] = acc0a[v]; Out[(size_t)r * FOUT + colb] = acc0b[v]; }
            r = m0 + 16 + mr; if (r < nrows) { Out[(size_t)r * FOUT + cola] = acc1a[v]; Out[(size_t)r * FOUT + colb] = acc1b[v]; }
            r = m0 + 32 + mr; if (r < nrows) { Out[(size_t)r * FOUT + cola] = acc2a[v]; Out[(size_t)r * FOUT + colb] = acc2b[v]; }
            r = m0 + 48 + mr; if (r < nrows) { Out[(size_t)r * FOUT + cola] = acc3a[v]; Out[(size_t)r * FOUT + colb] = acc3b[v]; }
        }
    }
}

// ---------------- el/er: per (node, head) dot with attention vectors ----------------
__global__ void el_er_kernel(const float* __restrict__ feat, const float* __restrict__ al,
                             const float* __restrict__ ar, float* __restrict__ el,
                             float* __restrict__ er, int nh) {
    int i = blockIdx.x * blockDim.x + threadIdx.x;
    if (i >= nh) return;
    const int node = i >> 2, h = i & 3;
    const float* f = feat + (size_t)node * FOUT + h * HDIM;
    const float* L = al + h * HDIM;
    const float* R = ar + h * HDIM;
    float sl = 0.f, sr = 0.f;
    #pragma unroll 8
    for (int d = 0; d < HDIM; ++d) { float fv = f[d]; sl += fv * L[d]; sr += fv * R[d]; }
    el[i] = sl; er[i] = sr;
}

// ---------------- edge pass 1: segment max of leaky_relu(el[src]+er[dst]) ----------------
__global__ void edge_max_kernel(const int* __restrict__ src, const int* __restrict__ dst,
                                const float* __restrict__ el, const float* __restrict__ er,
                                unsigned* __restrict__ emax_u, int eh) {
    int i = blockIdx.x * blockDim.x + threadIdx.x;
    if (i >= eh) return;
    const int e = i >> 2, h = i & 3;
    const int s = src[e], d = dst[e];
    float v = el[s * NHEAD + h] + er[d * NHEAD + h];
    float lr = v > 0.f ? v : NEGSLOPE * v;
    atomicMax(&emax_u[d * NHEAD + h], flipf(lr));
}

// ---------------- edge pass 2: ee = exp(e - emax[dst]); denom += ee ----------------
__global__ void edge_exp_kernel(const int* __restrict__ src, const int* __restrict__ dst,
                                const float* __restrict__ el, const float* __restrict__ er,
                                const unsigned* __restrict__ emax_u, float* __restrict__ ee,
                                float* __restrict__ denom, int eh) {
    int i = blockIdx.x * blockDim.x + threadIdx.x;
    if (i >= eh) return;
    const int e = i >> 2, h = i & 3;
    const int s = src[e], d = dst[e];
    float v = el[s * NHEAD + h] + er[d * NHEAD + h];
    float lr = v > 0.f ? v : NEGSLOPE * v;
    float m = unflipf(emax_u[d * NHEAD + h]);
    if (!__builtin_isfinite(m)) m = 0.f;     // matches jnp.where(isfinite, emax, 0)
    float ev = expf(lr - m);
    ee[i] = ev;
    atomicAdd(&denom[d * NHEAD + h], ev);
}

// ---------------- edge pass 3: rst[dst] += feat[src] * a ; 4 floats / thread ----------------
__global__ void edge_scatter_kernel(const int* __restrict__ src, const int* __restrict__ dst,
                                    const float* __restrict__ feat, const float* __restrict__ ee,
                                    const float* __restrict__ denom, float* __restrict__ rst) {
    int i = blockIdx.x * blockDim.x + threadIdx.x;     // over NEDGES*64
    if (i >= NEDGES * 64) return;
    const int e = i >> 6;
    const int r = i & 63;
    const int h = r >> 4;
    const int d0 = (r & 15) << 2;
    const int s = src[e], d = dst[e];
    float dn = denom[d * NHEAD + h];
    dn = dn > 1e-9f ? dn : 1e-9f;
    const float a = ee[e * NHEAD + h] / dn;
    const float4 f = *(const float4*)(feat + (size_t)s * FOUT + h * HDIM + d0);
    float* o = rst + (size_t)d * FOUT + h * HDIM + d0;
    atomicAdd(o + 0, f.x * a);
    atomicAdd(o + 1, f.y * a);
    atomicAdd(o + 2, f.z * a);
    atomicAdd(o + 3, f.w * a);
}

// ---------------- elu in place ----------------
__global__ void elu_kernel(float* __restrict__ p, int n) {
    int i = blockIdx.x * blockDim.x + threadIdx.x;
    if (i >= n) return;
    float x = p[i];
    p[i] = x > 0.f ? x : expm1f(x);
}

// ---------------- epilogue: transpose heads + per-head cluster softmax ----------------
// out[h][n][0:64] = h2[n][h][:] ; out[h][n][64:80] = softmax(h2[n][h] @ Wc + bc)
__global__ void final_kernel(const float* __restrict__ h2, const float* __restrict__ Wc,
                             const float* __restrict__ bc, float* __restrict__ out, int nh) {
    int i = blockIdx.x * blockDim.x + threadIdx.x;
    if (i >= nh) return;
    const int node = i >> 2, h = i & 3;
    const float* f = h2 + (size_t)node * FOUT + h * HDIM;
    float logits[NCLUS];
    #pragma unroll
    for (int c = 0; c < NCLUS; ++c) logits[c] = bc[c];
    for (int d = 0; d < HDIM; ++d) {
        float fv = f[d];
        const float* wrow = Wc + d * NCLUS;
        #pragma unroll
        for (int c = 0; c < NCLUS; ++c) logits[c] += fv * wrow[c];
    }
    float mx = logits[0];
    #pragma unroll
    for (int c = 1; c < NCLUS; ++c) mx = fmaxf(mx, logits[c]);
    float ssum = 0.f;
    #pragma unroll
    for (int c = 0; c < NCLUS; ++c) { logits[c] = expf(logits[c] - mx); ssum += logits[c]; }
    const float inv = 1.f / ssum;
    float* o = out + ((size_t)h * NNODES + node) * (HDIM + NCLUS);
    #pragma unroll 8
    for (int d = 0; d < HDIM; ++d) o[d] = f[d];
    #pragma unroll
    for (int c = 0; c < NCLUS; ++c) o[HDIM + c] = logits[c] * inv;
}

// ---------------- layer driver ----------------
static void run_layer(const float* in_feat, const float* Wmat, const float* al, const float* ar,
                      const int* src, const int* dst,
                      float* feat, float* rst, float* el, float* er,
                      unsigned* emax, float* denom, float* ee, hipStream_t stream) {
    const int NH  = NNODES * NHEAD;            // 200000
    const int EH  = NEDGES * NHEAD;            // 3200000
    const int NF  = NNODES * FOUT;             // 12800000
    const int LDS_BYTES = FIN * WLDSTRIDE * sizeof(float);   // 266,240

    // GEMM: feat = in_feat @ Wmat (grid-stride over M-tiles; 1 block per WGP due to LDS)
    {
        const int mtiles = (NNODES + 63) / 64;               // 782
        int blocks = mtiles < 256 ? mtiles : 256;
        gemm_f32_wmma<<<blocks, 256, LDS_BYTES, stream>>>(in_feat, Wmat, feat, NNODES, mtiles);
    }
    // el/er
    el_er_kernel<<<(NH + 255) / 256, 256, 0, stream>>>(feat, al, ar, el, er, NH);
    // init segment buffers (rst buffer is free by now: its prior contents were consumed)
    fill_u32<<<(NH + 255) / 256, 256, 0, stream>>>(emax, FLIP_NEG_INF, NH);
    fill_f32<<<(NH + 255) / 256, 256, 0, stream>>>(denom, 0.f, NH);
    fill_f32<<<(NF + 255) / 256, 256, 0, stream>>>(rst, 0.f, NF);
    // three edge sweeps
    edge_max_kernel<<<(EH + 255) / 256, 256, 0, stream>>>(src, dst, el, er, emax, EH);
    edge_exp_kernel<<<(EH + 255) / 256, 256, 0, stream>>>(src, dst, el, er, emax, ee, denom, EH);
    edge_scatter_kernel<<<(NEDGES * 64 + 255) / 256, 256, 0, stream>>>(src, dst, feat, ee, denom, rst);
    // activation (in place) -> layer output
    elu_kernel<<<(NF + 255) / 256, 256, 0, stream>>>(rst, NF);
}

extern "C" void kernel_launch(void* const* d_in, const int* in_sizes, int n_in,
                              void* d_out, int out_size, void* d_ws, size_t ws_size,
                              hipStream_t stream) {
    const float* x   = (const float*)d_in[0];
    const int*   src = (const int*)  d_in[1];
    const int*   dst = (const int*)  d_in[2];
    const float* W0  = (const float*)d_in[3];
    const float* al0 = (const float*)d_in[4];
    const float* ar0 = (const float*)d_in[5];
    const float* W1  = (const float*)d_in[6];
    const float* al1 = (const float*)d_in[7];
    const float* ar1 = (const float*)d_in[8];
    const float* Wc  = (const float*)d_in[9];
    const float* bc  = (const float*)d_in[10];
    float* out = (float*)d_out;

    // allow >64KB dynamic LDS for the GEMM (gfx1250 WGP has 320KB); ignore failure
    (void)hipFuncSetAttribute((const void*)gemm_f32_wmma,
                              hipFuncAttributeMaxDynamicSharedMemorySize,
                              FIN * WLDSTRIDE * (int)sizeof(float));

    // workspace carve-up (bytes, all 256-aligned)
    const size_t NB  = (size_t)NNODES * FOUT * sizeof(float);     // 51,200,000
    const size_t NH4 = (size_t)NNODES * NHEAD * sizeof(float);    //    800,000
    char* w = (char*)d_ws;
    float*    bufA  = (float*)(w);                 // feat (layer GEMM output)
    float*    bufB  = (float*)(w + NB);            // rst / h (layer output)
    float*    el    = (float*)(w + 2 * NB);
    float*    er    = (float*)(w + 2 * NB + NH4);
    unsigned* emax  = (unsigned*)(w + 2 * NB + 2 * NH4);
    float*    denom = (float*)(w + 2 * NB + 3 * NH4);
    float*    ee    = (float*)(w + 2 * NB + 4 * NH4);             // E*H floats = 12.8MB

    // layer 0: x -> feat(bufA) -> rst/h (bufB)
    run_layer(x, W0, al0, ar0, src, dst, bufA, bufB, el, er, emax, denom, ee, stream);
    // layer 1: h(bufB) -> feat(bufA) -> rst/h (bufB)  (gemm reads bufB before rst zero-fill)
    run_layer(bufB, W1, al1, ar1, src, dst, bufA, bufB, el, er, emax, denom, ee, stream);
    // epilogue: heads transpose + cluster softmax into d_out [H, N, D+C]
    final_kernel<<<(NNODES * NHEAD + 255) / 256, 256, 0, stream>>>(bufB, Wc, bc, out, NNODES * NHEAD);
}